// SOMLayer_32899449487566
// MI455X (gfx1250) — compile-verified
//
#include <hip/hip_runtime.h>

typedef float v2f __attribute__((ext_vector_type(2)));
typedef float v8f __attribute__((ext_vector_type(8)));
typedef int   v4i __attribute__((ext_vector_type(4)));

#define GLOBAL_V4I(p) ((__attribute__((address_space(1))) v4i*)(p))
#define LDS_V4I(p)    ((__attribute__((address_space(3))) v4i*)(p))

#define B_ 16
#define T_ 3072
#define D_ 256
#define G_ 1024
#define BT_ (B_ * T_)
#define ROWS 32
#define CCHUNK 64
#define NCHUNKS (G_ / CCHUNK)
#define ZS 260   /* padded row stride for z/node tiles: bank = (4*row + k) % 64, conflict-free */
#define QS 1028  /* padded q-buffer stride: float4-aligned, C-store conflict-free */

// LDS partition (float offsets)
#define OFF_ZT 0
#define OFF_NT0 (OFF_ZT + ROWS * ZS)      /* 8320  */
#define OFF_NT1 (OFF_NT0 + CCHUNK * ZS)   /* 24960 */
#define OFF_QB (OFF_NT1 + CCHUNK * ZS)    /* 41600 */
#define OFF_ZN (OFF_QB + ROWS * QS)       /* 74496 */
#define OFF_NN (OFF_ZN + ROWS)            /* 74528 */
#define OFF_MK (OFF_NN + G_)              /* 75552 */
#define OFF_IS (OFF_MK + ROWS)            /* 75584 */
#define OFF_BM (OFF_IS + ROWS)            /* 75616 */
#define SMEM_FLOATS (OFF_BM + ROWS)       /* 75648 floats = 302592 B < 320 KB */

__global__ __launch_bounds__(256, 1) void som_wmma_kernel(
    const float* __restrict__ z, const float* __restrict__ mask,
    const float* __restrict__ nodes,
    float* __restrict__ somz, float* __restrict__ qout,
    int* __restrict__ bmuOut, int* __restrict__ kcOut) {
  extern __shared__ float sm[];
  float* zT  = sm + OFF_ZT;
  float* nT0 = sm + OFF_NT0;
  float* nT1 = sm + OFF_NT1;
  float* qB  = sm + OFF_QB;
  float* zN  = sm + OFF_ZN;
  float* nN  = sm + OFF_NN;
  float* mk  = sm + OFF_MK;
  float* iS  = sm + OFF_IS;
  int*   bm  = (int*)(sm + OFF_BM);

  const int t = threadIdx.x;
  const int rowBase = blockIdx.x * ROWS;
  const float L2DECAY = -0.00144349547f;  // log2(0.999)

  // Async-DMA a 64-col node chunk into an LDS buffer (16 b128 ops/thread, ASYNCcnt += 16/wave)
  auto issueChunk = [&](int cgi, float* dstBuf) {
    const int colBase = cgi * CCHUNK;
#pragma unroll
    for (int i = 0; i < 16; ++i) {
      int idx = t + 256 * i;           // 4096 float4 chunks = 64 rows x 64 chunks
      int row = idx >> 6;
      int k4  = (idx & 63) << 2;
      __builtin_amdgcn_global_load_async_to_lds_b128(
          GLOBAL_V4I(nodes + (long)(colBase + row) * D_ + k4),
          LDS_V4I(dstBuf + row * ZS + k4),
          0, 0);
    }
  };

  issueChunk(0, nT0);  // chunk 0 in flight behind the z staging below

  // ---- stage weighted z tile (float4): zT[r][k] = z * decay^(T-1-tr) * mask ----
  for (int i = 0; i < 8; ++i) {
    int idx = t + 256 * i;             // 2048 float4 chunks = 32 rows x 64 chunks
    int r  = idx >> 6;
    int k4 = (idx & 63) << 2;
    int g  = rowBase + r;
    int tr = g % T_;
    float w = exp2f((float)(T_ - 1 - tr) * L2DECAY) * mask[g];
    float4 v = *(const float4*)(z + (long)g * D_ + k4);
    v.x *= w; v.y *= w; v.z *= w; v.w *= w;
    *(float4*)(zT + r * ZS + k4) = v;
  }
  if (t < ROWS) mk[t] = mask[rowBase + t];
  __syncthreads();

  // ---- z row squared norms (8 lanes per row, shfl reduce) ----
  {
    int r = t >> 3, sub = t & 7;
    const float* zp = zT + r * ZS + sub * 32;
    float s = 0.f;
    for (int k = 0; k < 32; ++k) { float v = zp[k]; s += v * v; }
    s += __shfl_xor(s, 1, 32);
    s += __shfl_xor(s, 2, 32);
    s += __shfl_xor(s, 4, 32);
    if (sub == 0) zN[r] = s;
  }

  // ---- wave tiling: 8 waves = 2 (row) x 4 (col) grid of 16x16 WMMA tiles ----
  const int wave = t >> 5, lane = t & 31;
  const int half = lane >> 4, l16 = lane & 15;
  const int wRow = (wave >> 2) * 16;
  const int wCol = (wave & 3) * 16;

  for (int cg = 0; cg < NCHUNKS; ++cg) {
    float* cur = (cg & 1) ? nT1 : nT0;
    float* nxt = (cg & 1) ? nT0 : nT1;
    const int colBase = cg * CCHUNK;

    // Prefetch next chunk, then wait until only its 16 async ops remain in flight
    // (async loads complete in order) -> current chunk is resident.
    if (cg < NCHUNKS - 1) {
      issueChunk(cg + 1, nxt);
      asm volatile("s_wait_asynccnt 0x10" ::: "memory");
    } else {
      asm volatile("s_wait_asynccnt 0x0" ::: "memory");
    }
    __syncthreads();

    // ---- node norms for this chunk, from LDS (4 lanes per col, shfl reduce) ----
    {
      int c = t >> 2, sub = t & 3;
      const float* p = cur + c * ZS + sub * 64;
      float s = 0.f;
      for (int k = 0; k < 64; ++k) { float v = p[k]; s += v * v; }
      s += __shfl_xor(s, 1, 32);
      s += __shfl_xor(s, 2, 32);
      if (sub == 0) nN[colBase + c] = s;
    }
    __syncthreads();

    // A: lane l16 = row M, VGPR pair holds K = {2*half, 2*half+1}; B symmetric with N.
    const float* aP = zT + (wRow + l16) * ZS + 2 * half;
    const float* bP = cur + (wCol + l16) * ZS + 2 * half;
    v8f acc = {};
#pragma unroll 8
    for (int k0 = 0; k0 < D_; k0 += 4) {
      v2f a = *(const v2f*)(aP + k0);
      v2f b = *(const v2f*)(bP + k0);
      acc = __builtin_amdgcn_wmma_f32_16x16x4_f32(
          /*neg_a=*/false, a, /*neg_b=*/false, b,
          /*c_mod=*/(short)0, acc, /*reuse_a=*/false, /*reuse_b=*/false);
    }
    // C/D layout: VGPR i, lanes 0-15 -> M=i, lanes 16-31 -> M=i+8; N = lane%16.
#pragma unroll
    for (int i = 0; i < 8; ++i) {
      int rl = wRow + i + 8 * half;
      int cgl = colBase + wCol + l16;
      float d2 = zN[rl] - 2.f * acc[i] + nN[cgl];
      float dist = __builtin_amdgcn_sqrtf(fmaxf(d2, 1e-12f));   // v_sqrt_f32
      qB[rl * QS + cgl] = __builtin_amdgcn_rcpf(1.f + dist);    // v_rcp_f32
    }
    __syncthreads();
  }

  // ---- per-row sum + argmax(q) (== argmin(dist), ties -> lowest index) ----
  {
    int r = t >> 3, sub = t & 7;
    float s = 0.f, bq = -1.f;
    int bi = 0;
    for (int c = sub; c < G_; c += 8) {
      float qv = qB[r * QS + c];
      s += qv;
      if (qv > bq) { bq = qv; bi = c; }
    }
    for (int m = 1; m <= 4; m <<= 1) {
      float s2 = __shfl_xor(s, m, 32);
      float q2 = __shfl_xor(bq, m, 32);
      int   i2 = __shfl_xor(bi, m, 32);
      s += s2;
      if (q2 > bq || (q2 == bq && i2 < bi)) { bq = q2; bi = i2; }
    }
    if (sub == 0) {
      iS[r] = __builtin_amdgcn_rcpf(s);
      bm[r] = bi;
      int g = rowBase + r;
      bmuOut[g] = bi;
      kcOut[2 * g]     = bi >> 5;  // bmu / GW
      kcOut[2 * g + 1] = bi & 31;  // bmu % GW
    }
  }
  __syncthreads();

  // ---- normalized q: ds_load_b128 + single HBM float4 store per thread per row ----
  for (int r = 0; r < ROWS; ++r) {
    float inv = iS[r];
    float4 v = *(const float4*)(qB + r * QS + 4 * t);  // QS % 4 == 0 -> aligned
    v.x *= inv; v.y *= inv; v.z *= inv; v.w *= inv;
    *(float4*)(qout + (long)(rowBase + r) * G_ + 4 * t) = v;
  }

  // ---- som_z = z + 0.1 * (nodes[bmu] - z) * mask ----
  for (int r = 0; r < ROWS; ++r) {
    int g = rowBase + r;
    float zv = z[(long)g * D_ + t];
    float nv = nodes[(long)bm[r] * D_ + t];
    somz[(long)g * D_ + t] = zv + 0.1f * (nv - zv) * mk[r];
  }
}

extern "C" void kernel_launch(void* const* d_in, const int* in_sizes, int n_in,
                              void* d_out, int out_size, void* d_ws, size_t ws_size,
                              hipStream_t stream) {
  const float* z     = (const float*)d_in[0];
  const float* mask  = (const float*)d_in[1];
  const float* nodes = (const float*)d_in[2];

  float* out  = (float*)d_out;
  float* somz = out;                                   // [B,T,D]   12,582,912 f32
  float* qout = somz + (size_t)BT_ * D_;               // [B*T,G]   50,331,648 f32
  int*   bmu  = (int*)(qout + (size_t)BT_ * G_);       // [B,T]     49,152 i32
  int*   kc   = bmu + (size_t)BT_;                     // [B,T,2]   98,304 i32

  size_t smem = (size_t)SMEM_FLOATS * sizeof(float);
  som_wmma_kernel<<<BT_ / ROWS, 256, smem, stream>>>(z, mask, nodes, somz, qout, bmu, kc);
}